// LightGCNConv_70188355551648
// MI455X (gfx1250) — compile-verified
//
#include <hip/hip_runtime.h>
#include <hip/hip_bf16.h>

typedef __attribute__((ext_vector_type(2))) float v2f;
typedef __attribute__((ext_vector_type(8))) float v8f;

#define D_FEAT      64
#define TILE_NODES  16
#define TILE_EDGES  256   // edge-metadata chunk staged in LDS (3 KB)
#define SCAN_T      1024

#if defined(__has_builtin)
#if __has_builtin(__builtin_amdgcn_global_load_async_to_lds_b32)
#define USE_ASYNC_LDS 1
#endif
#endif

#ifdef USE_ASYNC_LDS
// Toolchain's builtin signature takes generic int* params (per hipcc diag).
#define TO_I32P(p) ((int*)(uintptr_t)(p))
#endif

// ---------------- CSR build ----------------

__global__ void k_zero_i32(int* __restrict__ p, int n) {
    int i = blockIdx.x * blockDim.x + threadIdx.x;
    if (i < n) p[i] = 0;
}

__global__ void k_count(const int* __restrict__ edst, int* __restrict__ deg, int E) {
    int e = blockIdx.x * blockDim.x + threadIdx.x;
    if (e < E) atomicAdd(&deg[edst[e]], 1);
}

// single-workgroup exclusive scan: rp[0..N] = exclusive scan of deg[0..N);
// also seeds cursor[i] = rp[i].
__global__ void k_scan(const int* __restrict__ deg, int* __restrict__ rp,
                       int* __restrict__ cursor, int n) {
    __shared__ int sh[SCAN_T];
    __shared__ int carry_sh;
    const int tid = threadIdx.x;
    if (tid == 0) carry_sh = 0;
    __syncthreads();
    const int total = n + 1;
    for (int base = 0; base < total; base += SCAN_T) {
        const int i = base + tid;
        const int v = (i < n) ? deg[i] : 0;
        sh[tid] = v;
        __syncthreads();
        for (int off = 1; off < SCAN_T; off <<= 1) {
            int t = (tid >= off) ? sh[tid - off] : 0;
            __syncthreads();
            sh[tid] += t;
            __syncthreads();
        }
        const int incl = sh[tid];
        const int excl = carry_sh + incl - v;
        if (i <= n) {
            rp[i] = excl;
            if (i < n) cursor[i] = excl;
        }
        __syncthreads();
        if (tid == SCAN_T - 1) carry_sh += incl;
        __syncthreads();
    }
}

__global__ void k_fill(const int* __restrict__ esrc, const int* __restrict__ edst,
                       const float* __restrict__ ew, int* __restrict__ cursor,
                       int* __restrict__ csr_src, int* __restrict__ csr_dst,
                       float* __restrict__ csr_w, int E) {
    int e = blockIdx.x * blockDim.x + threadIdx.x;
    if (e >= E) return;
    const int d = edst[e];
    const int pos = atomicAdd(&cursor[d], 1);
    csr_src[pos] = esrc[e];
    csr_dst[pos] = d;
    csr_w[pos]   = ew[e];
}

// ---------------- WMMA SpMM layer ----------------
// One workgroup = 16 dst nodes x 64 features; 4 waves, wave w owns features
// [16w, 16w+16). Contiguous CSR edge range of the node tile is staged in LDS
// (async loads), padded to a multiple of 16 with sentinels, then consumed
// branch-free in groups of 16 edges = 4 x V_WMMA_F32_16X16X4_F32:
//   A[m,k] = w_k * (dst_k == node0+m)   (selection matrix, 2 VGPRs)
//   B[k,n] = hin[src_k][f0+n]           (gathered features, 2 VGPRs)
// Epilogue specialized at compile time:
//   FIRST:   acc = 0.25*(hin + C)   (fuses the x-init pass for layer 1)
//   !FIRST:  acc += 0.25*C
//   STORE_H: also write h_next = C  (skipped on the last layer)
template <int FIRST, int STORE_H>
__global__ void __launch_bounds__(128)
k_layer(const float* __restrict__ hin, const int* __restrict__ rp,
        const int* __restrict__ csr_src, const int* __restrict__ csr_dst,
        const float* __restrict__ csr_w, float* __restrict__ hout,
        float* __restrict__ acc, int N) {
    __shared__ __align__(16) int   s_src[TILE_EDGES];  // raw src node
    __shared__ __align__(16) int   s_dst[TILE_EDGES];  // raw dst node (-1 pad)
    __shared__ __align__(16) float s_w[TILE_EDGES];

    const int node0 = blockIdx.x * TILE_NODES;
    const int tid   = threadIdx.x;       // 0..127
    const int lane  = tid & 31;
    const int hi    = lane >> 4;         // 0 => K{0,1}, 1 => K{2,3}
    const int nlo   = lane & 15;         // M (for A) / N (for B,C)
    const int fofs  = ((tid >> 5) << 4) + nlo;   // feature column
    const int dmatch = node0 + nlo;      // dst value this lane selects on

    int top = node0 + TILE_NODES; if (top > N) top = N;
    const int pbeg = rp[node0];
    const int pend = rp[top];

    v8f c = {};
    for (int cbase = pbeg; cbase < pend; cbase += TILE_EDGES) {
        int cnt = pend - cbase; if (cnt > TILE_EDGES) cnt = TILE_EDGES;
        const int cntp = (cnt + 15) & ~15;

        // ---- stage metadata chunk into LDS (async), pad with sentinels ----
        for (int j = tid; j < cntp; j += 128) {
            if (j < cnt) {
                const int p = cbase + j;
#ifdef USE_ASYNC_LDS
                __builtin_amdgcn_global_load_async_to_lds_b32(
                    TO_I32P(csr_src + p), TO_I32P(s_src + j), 0, 0);
                __builtin_amdgcn_global_load_async_to_lds_b32(
                    TO_I32P(csr_dst + p), TO_I32P(s_dst + j), 0, 0);
                __builtin_amdgcn_global_load_async_to_lds_b32(
                    TO_I32P(csr_w + p),   TO_I32P(s_w + j),   0, 0);
#else
                s_src[j] = csr_src[p];
                s_dst[j] = csr_dst[p];
                s_w[j]   = csr_w[p];
#endif
            } else {
                s_src[j] = 0;
                s_dst[j] = -1;     // never matches any node id
                s_w[j]   = 0.0f;
            }
        }
#ifdef USE_ASYNC_LDS
#if __has_builtin(__builtin_amdgcn_s_wait_asynccnt)
        __builtin_amdgcn_s_wait_asynccnt(0);
#else
        asm volatile("s_wait_asynccnt 0x0" ::: "memory");
#endif
#endif
        __syncthreads();

        // ---- branch-free consume: 16 edges (4 WMMA K-steps) per iter ----
        for (int k = 0; k < cntp; k += 16) {
            v2f a[4], b[4];
#pragma unroll
            for (int u = 0; u < 4; ++u) {
                const int e = k + 4 * u + 2 * hi;   // even -> 8B aligned
                const int2 dd = *(const int2*)(s_dst + e);
                const v2f  ww = *(const v2f*)(s_w + e);
                const int2 ss = *(const int2*)(s_src + e);
                a[u].x = (dd.x == dmatch) ? ww.x : 0.0f;
                a[u].y = (dd.y == dmatch) ? ww.y : 0.0f;
                b[u].x = hin[(ss.x << 6) + fofs];
                b[u].y = hin[(ss.y << 6) + fofs];
            }
#pragma unroll
            for (int u = 0; u < 4; ++u)
                c = __builtin_amdgcn_wmma_f32_16x16x4_f32(
                        false, a[u], false, b[u], (short)0, c, false, false);
        }
        __syncthreads();
    }

    // C layout: VGPR r, lanes 0-15 -> (M=r, N=lane); lanes 16-31 -> (M=8+r, N=lane-16)
    const int idx0 = (node0 + 8 * hi) * D_FEAT + fofs;
    if (node0 + TILE_NODES <= N) {
        // full tile: batched, unguarded, fully specialized
        float av[8];
#pragma unroll
        for (int r = 0; r < 8; ++r)
            av[r] = FIRST ? hin[idx0 + r * D_FEAT] : acc[idx0 + r * D_FEAT];
#pragma unroll
        for (int r = 0; r < 8; ++r) {
            const int idx = idx0 + r * D_FEAT;
            if (STORE_H) hout[idx] = c[r];
            acc[idx] = FIRST ? 0.25f * (av[r] + c[r]) : (av[r] + 0.25f * c[r]);
        }
    } else {
#pragma unroll
        for (int r = 0; r < 8; ++r) {
            const int node = node0 + r + 8 * hi;
            if (node < N) {
                const int idx = node * D_FEAT + fofs;
                const float base = FIRST ? hin[idx] : acc[idx];
                if (STORE_H) hout[idx] = c[r];
                acc[idx] = FIRST ? 0.25f * (base + c[r]) : (base + 0.25f * c[r]);
            }
        }
    }
}

// ---------------- launch ----------------

static inline size_t align256(size_t x) { return (x + 255) & ~(size_t)255; }

extern "C" void kernel_launch(void* const* d_in, const int* in_sizes, int n_in,
                              void* d_out, int out_size, void* d_ws, size_t ws_size,
                              hipStream_t stream) {
    const float* x    = (const float*)d_in[0];
    const int*   esrc = (const int*)d_in[1];
    const int*   edst = (const int*)d_in[2];
    const float* ew   = (const float*)d_in[3];
    float*       out  = (float*)d_out;

    const int E = in_sizes[1];
    const int N = in_sizes[0] / D_FEAT;

    // carve workspace
    char* w = (char*)d_ws;
    size_t off = 0;
    int* deg      = (int*)(w + off); off = align256(off + (size_t)N * 4);
    int* rp       = (int*)(w + off); off = align256(off + (size_t)(N + 1) * 4);
    int* cursor   = (int*)(w + off); off = align256(off + (size_t)N * 4);
    int* csr_src  = (int*)(w + off); off = align256(off + (size_t)E * 4);
    int* csr_dst  = (int*)(w + off); off = align256(off + (size_t)E * 4);
    float* csr_w  = (float*)(w + off); off = align256(off + (size_t)E * 4);
    float* hA     = (float*)(w + off); off = align256(off + (size_t)N * D_FEAT * 4);
    float* hB     = (float*)(w + off); off = align256(off + (size_t)N * D_FEAT * 4);
    (void)ws_size; (void)n_in; (void)out_size;

    const int T = 256;
    // 1) CSR build
    k_zero_i32<<<(N + T - 1) / T, T, 0, stream>>>(deg, N);
    k_count<<<(E + T - 1) / T, T, 0, stream>>>(edst, deg, E);
    k_scan<<<1, SCAN_T, 0, stream>>>(deg, rp, cursor, N);
    k_fill<<<(E + T - 1) / T, T, 0, stream>>>(esrc, edst, ew, cursor,
                                              csr_src, csr_dst, csr_w, E);

    // 2) three propagation layers; out accumulates 0.25 * (x + h1 + h2 + h3)
    const int tiles = (N + TILE_NODES - 1) / TILE_NODES;
    k_layer<1, 1><<<tiles, 128, 0, stream>>>(x,  rp, csr_src, csr_dst, csr_w, hA, out, N);
    k_layer<0, 1><<<tiles, 128, 0, stream>>>(hA, rp, csr_src, csr_dst, csr_w, hB, out, N);
    k_layer<0, 0><<<tiles, 128, 0, stream>>>(hB, rp, csr_src, csr_dst, csr_w, hA, out, N);
}